// GeneralizedMahalanobis_68204080660525
// MI455X (gfx1250) — compile-verified
//
#include <hip/hip_runtime.h>
#include <hip/hip_bf16.h>

typedef __attribute__((ext_vector_type(2))) float v2f;
typedef __attribute__((ext_vector_type(8))) float v8f;

#define D_IN 128

// ---------------------------------------------------------------------------
// Phase 1: Y = X @ W  (fp32 WMMA, 16x16 output tile per wave, K chained by 4)
// Block = 256 threads = 8 waves; wave w owns column tile [16w, 16w+16).
// blockIdx.x owns row tile [16*bx, 16*bx+16). 128 cols = 8 waves exactly.
// ---------------------------------------------------------------------------
__global__ void __launch_bounds__(256)
gm_gemm_xw(const float* __restrict__ X, const float* __restrict__ W,
           float* __restrict__ Y, int Nrows) {
  const int lane = threadIdx.x & 31;
  const int wave = threadIdx.x >> 5;
  const int half = lane >> 4;   // 0: lanes 0-15, 1: lanes 16-31
  const int l16  = lane & 15;

  const int row0 = blockIdx.x * 16;
  const int col0 = wave * 16;

  // Clamp row for loads (no divergence: EXEC must stay all-1s around WMMA).
  int rA = row0 + l16;
  int rAc = rA < Nrows ? rA : (Nrows - 1);
  const float* xrow = X + (size_t)rAc * D_IN;

  v8f acc = {};
#pragma unroll
  for (int k0 = 0; k0 < D_IN; k0 += 4) {
    // A fragment (16x4): half 0 -> K = k0,k0+1 ; half 1 -> K = k0+2,k0+3
    const float* ap = xrow + k0 + half * 2;
    v2f a;
    a.x = ap[0];
    a.y = ap[1];
    // B fragment (4x16): half 0 -> rows k0,k0+1 ; half 1 -> rows k0+2,k0+3
    const float* bp = W + (size_t)(k0 + half * 2) * D_IN + col0 + l16;
    v2f b;
    b.x = bp[0];
    b.y = bp[D_IN];
    // 8 args: (neg_a, A, neg_b, B, c_mod, C, reuse_a, reuse_b)
    acc = __builtin_amdgcn_wmma_f32_16x16x4_f32(
        false, a, false, b, (short)0, acc, false, false);
  }

  // D layout: VGPR v -> row (row0 + half*8 + v), col (col0 + l16)
  const int rowD = row0 + half * 8;
#pragma unroll
  for (int v = 0; v < 8; ++v) {
    int r = rowD + v;
    if (r < Nrows) {
      Y[(size_t)r * D_IN + col0 + l16] = acc[v];
    }
  }
}

// ---------------------------------------------------------------------------
// Phase 2: per edge e: q = ||Y[i] - Y[j]||^2 ; out = exp(-sqrt(q)/2)
// One wave32 per edge; lane covers 4 contiguous floats (float4) of the row.
// Y (51.2 MB) is L2-resident on MI455X (192 MB L2) -> gathers are L2 traffic.
// ---------------------------------------------------------------------------
__global__ void __launch_bounds__(256)
gm_edge(const float* __restrict__ Y, const int* __restrict__ ei,
        const int* __restrict__ ej, float* __restrict__ out, int E) {
  const int lane = threadIdx.x & 31;
  const int wave = threadIdx.x >> 5;
  const int e = blockIdx.x * 8 + wave;
  if (e >= E) return;

  const int i = ei[e];
  const int j = ej[e];

  const float4 a = ((const float4*)(Y + (size_t)i * D_IN))[lane];
  const float4 b = ((const float4*)(Y + (size_t)j * D_IN))[lane];

  const float dx = a.x - b.x;
  const float dy = a.y - b.y;
  const float dz = a.z - b.z;
  const float dw = a.w - b.w;
  float q = dx * dx + dy * dy + dz * dz + dw * dw;

  // wave32 butterfly reduction
#pragma unroll
  for (int off = 16; off > 0; off >>= 1) {
    q += __shfl_xor(q, off, 32);
  }

  if (lane == 0) {
    out[e] = expf(-sqrtf(q) * 0.5f);  // SIGMA=1 -> exp(-dist/2)
  }
}

// ---------------------------------------------------------------------------
extern "C" void kernel_launch(void* const* d_in, const int* in_sizes, int n_in,
                              void* d_out, int out_size, void* d_ws, size_t ws_size,
                              hipStream_t stream) {
  const float* X    = (const float*)d_in[0];   // [N, 128]
  const float* W    = (const float*)d_in[1];   // [128, 128]
  const int*   edge = (const int*)d_in[2];     // [2, E] (int32 per harness)
  float* out = (float*)d_out;

  const int N = in_sizes[0] / D_IN;
  const int E = in_sizes[2] / 2;

  float* Y = (float*)d_ws;                     // N*128 floats = 51.2 MB scratch

  // Phase 1: Y = X @ W via fp32 WMMA
  gm_gemm_xw<<<(N + 15) / 16, 256, 0, stream>>>(X, W, Y, N);

  // Phase 2: edge distances (8 edges per 256-thread block, 1 wave/edge)
  gm_edge<<<(E + 7) / 8, 256, 0, stream>>>(Y, edge, edge + E, out, E);
}